// Pre_ProcessLayer_Graph_35081292873880
// MI455X (gfx1250) — compile-verified
//
#include <hip/hip_runtime.h>
#include <hip/hip_bf16.h>
#include <float.h>
#include <limits.h>

// Problem constants (match reference)
#define BB    4
#define CIN   3
#define HIN   128
#define WIN   128
#define COUT  64
#define KK    3
#define HOUT  64
#define WOUT  64
#define NN    (HOUT * WOUT)        // 4096 nodes per batch
#define TOPK  7                    // NUM_NEIGHBORS

typedef __attribute__((ext_vector_type(2))) float v2f;
typedef __attribute__((ext_vector_type(8))) float v8f;

// ---------------------------------------------------------------------------
// Kernel 1: 3x3 stride-2 conv (pad 1) + bias, output in [B, N, C] node layout.
// ---------------------------------------------------------------------------
__global__ void conv_head_kernel(const float* __restrict__ x,
                                 const float* __restrict__ wgt,
                                 const float* __restrict__ bias,
                                 float* __restrict__ y) {
    int idx = blockIdx.x * blockDim.x + threadIdx.x;   // [0, B*COUT*H*W)
    int wo = idx & (WOUT - 1);
    int ho = (idx >> 6) & (HOUT - 1);
    int co = (idx >> 12) & (COUT - 1);
    int b  = idx >> 18;

    float acc = bias[co];
    #pragma unroll
    for (int ci = 0; ci < CIN; ++ci) {
        #pragma unroll
        for (int kh = 0; kh < KK; ++kh) {
            int ih = 2 * ho - 1 + kh;
            if (ih < 0 || ih >= HIN) continue;
            #pragma unroll
            for (int kw = 0; kw < KK; ++kw) {
                int iw = 2 * wo - 1 + kw;
                if (iw < 0 || iw >= WIN) continue;
                float xv = x[(((size_t)b * CIN + ci) * HIN + ih) * WIN + iw];
                float wv = wgt[((co * CIN + ci) * KK + kh) * KK + kw];
                acc = fmaf(xv, wv, acc);
            }
        }
    }
    int n = ho * WOUT + wo;
    y[((size_t)b * NN + n) * COUT + co] = acc;   // [B, N, C]
}

// ---------------------------------------------------------------------------
// Kernel 2: per-node squared norms sq[b][n] = sum_c y^2
// ---------------------------------------------------------------------------
__global__ void sqnorm_kernel(const float* __restrict__ y,
                              float* __restrict__ sq) {
    int idx = blockIdx.x * blockDim.x + threadIdx.x;   // [0, B*N)
    const float* row = y + (size_t)idx * COUT;
    float s = 0.f;
    #pragma unroll
    for (int c = 0; c < COUT; ++c) s = fmaf(row[c], row[c], s);
    sq[idx] = s;
}

// ---------------------------------------------------------------------------
// Kernel 3: WMMA f32 Gram matmul -> d2 = max(sq_n + sq_m - 2*y.yT, 0)
// One wave per 16x64 strip of d2 (4 N-tiles): A operand loaded once per
// K-step and reused across 4 WMMAs -> 1.25 loads/WMMA instead of 2.0.
// K-loop: 16 steps x 4 x V_WMMA_F32_16X16X4_F32 (exact f32 tensor path).
// d2 goes straight into the adjacency region of d_out (rewritten in place
// by the top-k kernel) to stay at the HBM-traffic floor.
//
// ISA layouts (wave32):
//   A 16x4 f32 : lane[3:0] = M row, VGPR v holds K = 2*lane[4] + v
//   B 4x16 f32 : lane[3:0] = N col, VGPR v holds K = 2*lane[4] + v
//   -> with row-major y, A and B operands are identical float2 loads.
//   C/D 16x16  : VGPR i -> M = i + 8*lane[4], N = lane[3:0]
// ---------------------------------------------------------------------------
__global__ void gram_d2_kernel(const float* __restrict__ y,
                               const float* __restrict__ sq,
                               float* __restrict__ d2) {
    const int wavesPerBlock = blockDim.x >> 5;
    const int wave = blockIdx.x * wavesPerBlock + (threadIdx.x >> 5);
    const int lane = threadIdx.x & 31;

    const int tj4 = wave & 63;          // strip of 4 N-tiles (columns m)
    const int ti  = (wave >> 6) & 255;  // M-tile (rows n)
    const int b   = wave >> 14;         // batch

    const int laneRow = lane & 15;          // M (for A) / N (for B) within tile
    const int kSel    = (lane >> 4) << 1;   // 0 or 2: which K pair this lane holds

    const float* yb   = y + (size_t)b * NN * COUT;
    const float* aRow = yb + (size_t)(ti  * 16 + laneRow) * COUT;
    const float* bRow = yb + (size_t)(tj4 * 64 + laneRow) * COUT;

    v8f acc0 = {}, acc1 = {}, acc2 = {}, acc3 = {};
    #pragma unroll
    for (int s = 0; s < 16; ++s) {
        const int kb = s * 4 + kSel;
        v2f av = *(const v2f*)(aRow + kb);                 // shared A operand
        v2f b0 = *(const v2f*)(bRow + kb);                 // 4 B tiles: +4KB strides
        v2f b1 = *(const v2f*)(bRow + kb + 16 * COUT);     //   (fit 24-bit ioffset)
        v2f b2 = *(const v2f*)(bRow + kb + 32 * COUT);
        v2f b3 = *(const v2f*)(bRow + kb + 48 * COUT);
        acc0 = __builtin_amdgcn_wmma_f32_16x16x4_f32(false, av, false, b0,
                                                     (short)0, acc0, false, false);
        acc1 = __builtin_amdgcn_wmma_f32_16x16x4_f32(false, av, false, b1,
                                                     (short)0, acc1, false, false);
        acc2 = __builtin_amdgcn_wmma_f32_16x16x4_f32(false, av, false, b2,
                                                     (short)0, acc2, false, false);
        acc3 = __builtin_amdgcn_wmma_f32_16x16x4_f32(false, av, false, b3,
                                                     (short)0, acc3, false, false);
    }

    // Epilogue: d2 = max(sq[n] + sq[m] - 2*G, 0)
    const int nCol  = lane & 15;                    // N within tile
    const int mBase = ti * 16 + ((lane >> 4) << 3); // first of my 8 M rows
    float sqn[8];
    #pragma unroll
    for (int i = 0; i < 8; ++i) sqn[i] = sq[b * NN + mBase + i];

    const size_t adjBase = (size_t)b * NN * NN;
    v8f accs[4] = {acc0, acc1, acc2, acc3};
    #pragma unroll
    for (int j = 0; j < 4; ++j) {
        const int mCol = tj4 * 64 + j * 16 + nCol;  // global column index
        const float sqm = sq[b * NN + mCol];
        #pragma unroll
        for (int i = 0; i < 8; ++i) {
            float v = sqn[i] + sqm - 2.0f * accs[j][i];
            v = fmaxf(v, 0.0f);                     // matches reference tie behavior
            d2[adjBase + (size_t)(mBase + i) * NN + mCol] = v;
        }
    }
}

// ---------------------------------------------------------------------------
// Kernel 4: per-row top-7 (stable by (value,index), matching double-argsort),
// rewriting the d2 row in place with the 0/1 adjacency.
// One 256-thread workgroup per row; each thread handles 16 elements.
// ---------------------------------------------------------------------------
__global__ void topk_adj_kernel(float* __restrict__ adj) {
    __shared__ float cv[256 * TOPK];
    __shared__ int   ci[256 * TOPK];
    __shared__ int   gidx[TOPK];

    const int t = threadIdx.x;
    float* row = adj + (size_t)blockIdx.x * NN;

    // Load my 16 values, keep local stable top-7 by (value, index).
    float lv[TOPK];
    int   li[TOPK];
    #pragma unroll
    for (int k = 0; k < TOPK; ++k) { lv[k] = FLT_MAX; li[k] = INT_MAX; }

    #pragma unroll
    for (int j = 0; j < 16; ++j) {
        const int idx = t * 16 + j;
        const float v = row[idx];
        if (v < lv[TOPK - 1] || (v == lv[TOPK - 1] && idx < li[TOPK - 1])) {
            int p = TOPK - 1;
            while (p > 0 && (v < lv[p - 1] || (v == lv[p - 1] && idx < li[p - 1]))) {
                lv[p] = lv[p - 1]; li[p] = li[p - 1]; --p;
            }
            lv[p] = v; li[p] = idx;
        }
    }
    #pragma unroll
    for (int k = 0; k < TOPK; ++k) { cv[t * TOPK + k] = lv[k]; ci[t * TOPK + k] = li[k]; }
    __syncthreads();

    // Thread 0 merges 256*7 candidates into the global stable top-7.
    if (t == 0) {
        float gv[TOPK];
        int   gi[TOPK];
        #pragma unroll
        for (int k = 0; k < TOPK; ++k) { gv[k] = FLT_MAX; gi[k] = INT_MAX; }
        for (int c = 0; c < 256 * TOPK; ++c) {
            const float v = cv[c];
            const int   id = ci[c];
            if (v < gv[TOPK - 1] || (v == gv[TOPK - 1] && id < gi[TOPK - 1])) {
                int p = TOPK - 1;
                while (p > 0 && (v < gv[p - 1] || (v == gv[p - 1] && id < gi[p - 1]))) {
                    gv[p] = gv[p - 1]; gi[p] = gi[p - 1]; --p;
                }
                gv[p] = v; gi[p] = id;
            }
        }
        #pragma unroll
        for (int k = 0; k < TOPK; ++k) gidx[k] = gi[k];
    }
    __syncthreads();

    // Rewrite my 16 elements with 0/1.
    int top[TOPK];
    #pragma unroll
    for (int k = 0; k < TOPK; ++k) top[k] = gidx[k];
    #pragma unroll
    for (int j = 0; j < 16; ++j) {
        const int idx = t * 16 + j;
        float o = 0.0f;
        #pragma unroll
        for (int k = 0; k < TOPK; ++k) if (idx == top[k]) o = 1.0f;
        row[idx] = o;
    }
}

// ---------------------------------------------------------------------------
extern "C" void kernel_launch(void* const* d_in, const int* in_sizes, int n_in,
                              void* d_out, int out_size, void* d_ws, size_t ws_size,
                              hipStream_t stream) {
    const float* x    = (const float*)d_in[0];   // [4,3,128,128]
    const float* wgt  = (const float*)d_in[1];   // [64,3,3,3]
    const float* bias = (const float*)d_in[2];   // [64]

    float* yOut = (float*)d_out;                         // [4,4096,64]
    float* adj  = yOut + (size_t)BB * NN * COUT;         // [4,4096,4096]
    float* sq   = (float*)d_ws;                          // [4,4096] = 64 KB scratch

    // 1) conv head -> y (node features)
    {
        const int total = BB * COUT * HOUT * WOUT;       // 1,048,576
        conv_head_kernel<<<total / 256, 256, 0, stream>>>(x, wgt, bias, yOut);
    }
    // 2) squared norms
    {
        const int total = BB * NN;                       // 16,384
        sqnorm_kernel<<<total / 256, 256, 0, stream>>>(yOut, sq);
    }
    // 3) WMMA Gram matmul -> d2 written into adjacency region
    {
        const int strips = BB * (NN / 16) * (NN / 64);   // 65,536 waves (16x64 strips)
        const int wavesPerBlock = 8;                     // 256 threads
        gram_d2_kernel<<<strips / wavesPerBlock, wavesPerBlock * 32, 0, stream>>>(yOut, sq, adj);
    }
    // 4) per-row stable top-7 -> 0/1 adjacency (in place)
    {
        topk_adj_kernel<<<BB * NN, 256, 0, stream>>>(adj);
    }
}